// SeparateHiddenDecoder_26800595927060
// MI455X (gfx1250) — compile-verified
//
#include <hip/hip_runtime.h>

typedef __attribute__((ext_vector_type(2))) float v2f;
typedef __attribute__((ext_vector_type(8))) float v8f;

// ---------------- degree / normalization ----------------

__global__ void fill_ones_kernel(float* __restrict__ p, int n) {
    int i = blockIdx.x * blockDim.x + threadIdx.x;
    if (i < n) p[i] = 1.0f;   // self-loop contributes 1 to every node's degree
}

__global__ void deg_accum_kernel(float* __restrict__ deg, const int* __restrict__ dst, int E) {
    int e = blockIdx.x * blockDim.x + threadIdx.x;
    if (e < E) atomicAdd(&deg[dst[e]], 1.0f);
}

__global__ void deg_to_dinv_kernel(float* __restrict__ d, int n) {
    int i = blockIdx.x * blockDim.x + threadIdx.x;
    if (i < n) {
        float v = d[i];
        d[i] = (v > 0.0f) ? rsqrtf(v) : 0.0f;
    }
}

// ---------------- fp32 WMMA GEMM: Y[N,OUT] = X[N,128] @ W[128,OUT] ----------------
// 256 threads = 8 waves; each wave computes a 16-row stripe slice x OUT cols.
// Persistent blocks grid-stride over 128-row stripes; W is staged once per block
// into LDS in a pair-interleaved layout so every B fragment is a single
// ds_load_b64 off ONE per-lane base address with immediate offsets:
//   W[k][n] -> wt[(k>>1)*(2*OUT) + 2*n + (k&1)]
// Fragment (W[kb][col], W[kb+1][col]) lives at float index
//   (k>>1 + half)*2*OUT + 2*col   (kb = k + 2*half, col = m + 16*t)
// => byte offset from lane base: (k>>1)*8*OUT + 128*t   (max 64,384 < 2^16)

template <int NT>   // OUT = NT*16
__global__ __launch_bounds__(256) void gemm_wmma_kernel(const float* __restrict__ X,
                                                        const float* __restrict__ W,
                                                        float* __restrict__ Y,
                                                        int N, int numStripes) {
    constexpr int K = 128;
    constexpr int OUT = NT * 16;
    extern __shared__ float wt[];           // K*OUT floats, pair-interleaved

    const int tid = threadIdx.x;
    for (int i = tid; i < K * OUT; i += 256) {
        int k = i / OUT;
        int n = i - k * OUT;
        wt[(k >> 1) * (2 * OUT) + 2 * n + (k & 1)] = W[i];   // W row-major [K][OUT]
    }
    __syncthreads();

    const int lane = tid & 31;
    const int wave = tid >> 5;
    const int m = lane & 15;                // A: row in tile / B,D: col in tile
    const int half = lane >> 4;             // selects K-pair group (A/B) or M+8 (D)
    const int kg = half << 1;               // 0 or 2

    const float* lbase = wt + (half * OUT + m) * 2;   // per-lane LDS base for B frags

    for (int stripe = blockIdx.x; stripe < numStripes; stripe += gridDim.x) {
        const int rowBase = stripe * 128 + wave * 16;
        int arow = rowBase + m;
        if (arow >= N) arow = N - 1;        // clamp; store is guarded, lanes stay active
        const float* xrow = X + (long)arow * K;

        v8f zero = {};
        v8f acc[NT];
#pragma unroll
        for (int t = 0; t < NT; ++t) acc[t] = zero;

#pragma unroll 8
        for (int k = 0; k < K; k += 4) {
            // A fragment (16x4 f32): lane holds X[row][k+kg], X[row][k+kg+1]
            v2f a = *(const v2f*)(xrow + k + kg);
            // All NT B fragments first -> one dscnt wait covers NT WMMAs
            v2f b[NT];
            const float* lk = lbase + (k >> 1) * (2 * OUT);
#pragma unroll
            for (int t = 0; t < NT; ++t) b[t] = *(const v2f*)(lk + t * 32);
#pragma unroll
            for (int t = 0; t < NT; ++t)
                acc[t] = __builtin_amdgcn_wmma_f32_16x16x4_f32(
                    false, a, false, b[t], (short)0, acc[t], false, false);
        }

        // D layout: VGPR r -> M = r + 8*half, N = m
#pragma unroll
        for (int t = 0; t < NT; ++t) {
            const int col = t * 16 + m;
#pragma unroll
            for (int r = 0; r < 8; ++r) {
                const int row = rowBase + r + 8 * half;
                if (row < N) Y[(long)row * OUT + col] = acc[t][r];
            }
        }
    }
}

// ---------------- self-loop + bias init: out = b + xw * dinv^2 ----------------

__global__ void self_bias_kernel(float* __restrict__ out, const float* __restrict__ xw,
                                 const float* __restrict__ b, const float* __restrict__ dinv,
                                 int N, int DIM) {
    long idx = (long)blockIdx.x * blockDim.x + threadIdx.x;
    if (idx >= (long)N * DIM) return;
    int i = (int)(idx / DIM);
    int c = (int)(idx - (long)i * DIM);
    float dv = dinv[i];
    out[idx] = b[c] + xw[idx] * dv * dv;
}

// ---------------- edge scatter: out[dst] += xw[src] * dinv[src]*dinv[dst] ----------------
// One wave per edge; each lane handles VPL consecutive channels (coalesced 512B/256B gather,
// f32 atomics resolve in the 192MB L2 where xw/out are resident).

template <int VPL>  // DIM = 32*VPL
__global__ __launch_bounds__(256) void edge_scatter_kernel(const float* __restrict__ xw,
                                                           const float* __restrict__ dinv,
                                                           const int* __restrict__ src,
                                                           const int* __restrict__ dst,
                                                           float* __restrict__ out, int E) {
    constexpr int DIM = 32 * VPL;
    long g = (long)blockIdx.x * blockDim.x + threadIdx.x;
    int e = (int)(g >> 5);
    int lane = (int)(g & 31);
    if (e >= E) return;
    int s = src[e];
    int d = dst[e];
    float w = dinv[s] * dinv[d];
    const float* p = xw + (long)s * DIM + lane * VPL;
    float* q = out + (long)d * DIM + lane * VPL;
    float v[VPL];
#pragma unroll
    for (int i = 0; i < VPL; ++i) v[i] = p[i];
#pragma unroll
    for (int i = 0; i < VPL; ++i) atomicAdd(q + i, v[i] * w);
}

// ---------------- launch ----------------

extern "C" void kernel_launch(void* const* d_in, const int* in_sizes, int n_in,
                              void* d_out, int out_size, void* d_ws, size_t ws_size,
                              hipStream_t stream) {
    const float* x  = (const float*)d_in[0];
    const int*   ei = (const int*)d_in[1];     // [2, E] flat: row0=src, row1=dst (int32)
    const float* W1 = (const float*)d_in[2];
    const float* b1 = (const float*)d_in[3];
    const float* W2 = (const float*)d_in[4];
    const float* b2 = (const float*)d_in[5];
    float* out = (float*)d_out;

    const int N = in_sizes[0] / 128;
    const int E = in_sizes[1] / 2;
    const int* src = ei;
    const int* dst = ei + E;

    // workspace layout: dinv[N] | xw[N*128] | h[N*128]; layer-2 xw reuses xw slot
    float* dinv = (float*)d_ws;
    float* xw   = dinv + N;
    float* h    = xw + (size_t)N * 128;

    // 1) degrees (shared across both layers) -> dinv
    fill_ones_kernel<<<(N + 255) / 256, 256, 0, stream>>>(dinv, N);
    deg_accum_kernel<<<(E + 255) / 256, 256, 0, stream>>>(dinv, dst, E);
    deg_to_dinv_kernel<<<(N + 255) / 256, 256, 0, stream>>>(dinv, N);

    const int numStripes = (N + 127) / 128;
    const int gemmBlocks = numStripes < 320 ? numStripes : 320;  // persistent blocks
    const long et = (long)E * 32;

    // ---- layer 1: 128 -> 128 ----
    gemm_wmma_kernel<8><<<gemmBlocks, 256, 128 * 128 * sizeof(float), stream>>>(x, W1, xw, N, numStripes);
    {
        long tot = (long)N * 128;
        self_bias_kernel<<<(int)((tot + 255) / 256), 256, 0, stream>>>(h, xw, b1, dinv, N, 128);
    }
    edge_scatter_kernel<4><<<(int)((et + 255) / 256), 256, 0, stream>>>(xw, dinv, src, dst, h, E);

    // ---- layer 2: 128 -> 64 ----
    gemm_wmma_kernel<4><<<gemmBlocks, 256, 64 * 128 * sizeof(float), stream>>>(h, W2, xw, N, numStripes);
    {
        long tot = (long)N * 64;
        self_bias_kernel<<<(int)((tot + 255) / 256), 256, 0, stream>>>(out, xw, b2, dinv, N, 64);
    }
    edge_scatter_kernel<2><<<(int)((et + 255) / 256), 256, 0, stream>>>(xw, dinv, src, dst, out, E);
}